// AdaptiveCentralLayer3D_12936441495587
// MI455X (gfx1250) — compile-verified
//
#include <hip/hip_runtime.h>
#include <stdint.h>

// AdaptiveCentralLayer3D for MI455X (gfx1250):
//   B=16, C=64, D=H=W=32, fp32.  1024 slices of 128 KB each.
//   Single HBM pass (memory-bound, floor ~11.5us @ 23.3 TB/s):
//     async-stage slice to LDS -> centroid reduction -> rolled gather
//     LDS->global via async from-LDS stores (per-lane LDS source address).

#define SLICE   32768                     // 32*32*32 elements per (b,c)
#define THREADS 512                       // 16 wave32 waves; 2 blocks/WGP (LDS)
#define CHUNKS  (SLICE / (THREADS * 4))   // 16 async B128 loads per thread
#define OUT_IT  (SLICE / THREADS)         // 64 output elements per thread

// Stage-in: one base global VGPR pair + one base LDS VGPR, per-load literal
// IOFFSET (HW adds the immediate to BOTH the LDS and the global address).
template<int J>
struct StageIn {
    static __device__ __forceinline__ void go(uint32_t la, uint64_t ga) {
        asm volatile("global_load_async_to_lds_b128 %0, %1, off offset:%2 th:TH_LOAD_NT"
                     :: "v"(la), "v"(ga), "i"(J * THREADS * 16) : "memory");
        StageIn<J + 1>::go(la, ga);
    }
};
template<> struct StageIn<CHUNKS> {
    static __device__ __forceinline__ void go(uint32_t, uint64_t) {}
};

__global__ __launch_bounds__(THREADS)
void center3d_kernel(const float* __restrict__ in, float* __restrict__ out)
{
    extern __shared__ float lds[];          // [SLICE] tile + [64] wave partials
    float* tile = lds;
    float* red  = lds + SLICE;

    const int tid  = threadIdx.x;
    const int wave = tid >> 5;              // wave32
    const uint64_t slice_base = (uint64_t)blockIdx.x * (uint64_t)SLICE;
    const float* gin = in + slice_base;

    // LDS byte offset of tile[0]: low 32 bits of the generic pointer are the
    // LDS offset (aperture lives in addr[63:32]).
    const uint32_t tile_off = (uint32_t)(uintptr_t)tile;

    // ---------------- Phase 1: async global -> LDS ---------------------------
    // Thread covers elements (j*THREADS + tid)*4 .. +3;  16 B/lane/op,
    // 512 B contiguous per wave.  NT: streamed once, don't retain in L2.
    StageIn<0>::go(tile_off + (uint32_t)tid * 16u,
                   (uint64_t)(uintptr_t)(gin + tid * 4));
    asm volatile("s_wait_asynccnt 0" ::: "memory");   // own wave's loads resident

    // ---------------- Reduction: Sum k, Sum k*z, Sum k*y, Sum k*x ------------
    float ak = 0.f, az = 0.f, ay = 0.f, ax = 0.f;
    #pragma unroll 4
    for (int j = 0; j < CHUNKS; ++j) {
        const int v4   = j * THREADS + tid;
        const float4 v = reinterpret_cast<const float4*>(tile)[v4];  // ds_load_b128
        const int f    = v4 * 4;
        const float d  = (float)(f >> 10);
        const float h  = (float)((f >> 5) & 31);
        const float w  = (float)(f & 31);             // w..w+3 within one row
        const float s  = v.x + v.y + v.z + v.w;
        ak += s;
        az += d * s;
        ay += h * s;
        ax += w * s + (v.y + 2.f * v.z + 3.f * v.w);
    }
    #pragma unroll
    for (int m = 16; m > 0; m >>= 1) {                // wave32 butterfly
        ak += __shfl_xor(ak, m, 32);
        az += __shfl_xor(az, m, 32);
        ay += __shfl_xor(ay, m, 32);
        ax += __shfl_xor(ax, m, 32);
    }
    if ((tid & 31) == 0) {
        red[wave * 4 + 0] = ak;
        red[wave * 4 + 1] = az;
        red[wave * 4 + 2] = ay;
        red[wave * 4 + 3] = ax;
    }
    __syncthreads();   // also: ALL waves' async-loaded tile data now visible

    // Every thread folds the 16 wave partials in a fixed order (deterministic).
    float tk = 0.f, tz = 0.f, ty = 0.f, tx = 0.f;
    #pragma unroll
    for (int wv = 0; wv < THREADS / 32; ++wv) {
        tk += red[wv * 4 + 0];
        tz += red[wv * 4 + 1];
        ty += red[wv * 4 + 2];
        tx += red[wv * 4 + 3];
    }
    const float inv = 1.f / (tk + 1e-8f);
    // shift = round-half-to-even(D/2 - centroid); rintf under default RNE mode
    const int sz = (int)rintf(16.f - tz * inv);
    const int sy = (int)rintf(16.f - ty * inv);
    const int sx = (int)rintf(16.f - tx * inv);

    float* gout = out + slice_base;

    // ------------- Phase 2: rolled gather LDS -> global (async B32) ----------
    // out[f] = tile[rolled(f)]; per-lane LDS source address fused into the
    // store.  w-source is loop-invariant (w = tid&31); row term is
    // wave-uniform.  (i - s) & 31 is a correct mod-32 even for negative i-s.
    const int ws = ((tid & 31) - sx) & 31;
    const uint32_t lane_lds = tile_off + ((uint32_t)ws << 2);
    uint64_t ga = (uint64_t)(uintptr_t)(gout + tid);
    #pragma unroll 4
    for (int i = 0; i < OUT_IT; ++i) {
        const int f = i * THREADS + tid;
        const int h = (f >> 5) & 31, d = f >> 10;
        const uint32_t row = (uint32_t)((((d - sz) & 31) << 10)
                                      | (((h - sy) & 31) << 5)) << 2;
        const uint32_t la = lane_lds + row;
        asm volatile("global_store_async_from_lds_b32 %0, %1, off th:TH_STORE_NT"
                     :: "v"(ga), "v"(la) : "memory");
        ga += (uint64_t)THREADS * 4u;
    }
    asm volatile("s_wait_asynccnt 0" ::: "memory");   // drain before endpgm
}

extern "C" void kernel_launch(void* const* d_in, const int* in_sizes, int n_in,
                              void* d_out, int out_size, void* d_ws, size_t ws_size,
                              hipStream_t stream)
{
    (void)n_in; (void)d_ws; (void)ws_size; (void)out_size;
    const float* in  = (const float*)d_in[0];
    float*       out = (float*)d_out;

    const int nslices = in_sizes[0] / SLICE;                    // 16*64 = 1024
    const size_t smem = (size_t)(SLICE + 64) * sizeof(float);   // 128 KB + pad

    // Allow >64 KB dynamic LDS (WGP supports 320 KB). Idempotent, host-side.
    hipFuncSetAttribute((const void*)center3d_kernel,
                        hipFuncAttributeMaxDynamicSharedMemorySize, (int)smem);

    center3d_kernel<<<dim3(nslices), dim3(THREADS), smem, stream>>>(in, out);
}